// GDIFNet_59682865545266
// MI455X (gfx1250) — compile-verified
//
#include <hip/hip_runtime.h>
#include <math.h>

// ---------------------------------------------------------------------------
// GDIFNet forward for MI455X (gfx1250). fp32 end-to-end; GAT projections use
// V_WMMA_F32_16X16X4_F32. Edge phase (gather + atomic scatter) is the
// bandwidth-dominant part; all scratch (~78 MB) fits in the 192 MB L2.
// ---------------------------------------------------------------------------

#define DLAT 64
#define MMOD 4
#define NHEAD 4
#define NLAY 2
#define OD (NHEAD * DLAT) // 256

typedef float v2f __attribute__((ext_vector_type(2)));
typedef float v8f __attribute__((ext_vector_type(8)));

__device__ __forceinline__ float gelu_exact(float x) {
    return 0.5f * x * (1.0f + erff(x * 0.70710678118654752f));
}
// order-preserving float -> uint encoding for atomic max
__device__ __forceinline__ unsigned fenc(float f) {
    unsigned u = __float_as_uint(f);
    return (u & 0x80000000u) ? ~u : (u | 0x80000000u);
}
__device__ __forceinline__ float fdec(unsigned u) {
    unsigned b = (u & 0x80000000u) ? (u & 0x7FFFFFFFu) : ~u;
    return __uint_as_float(b);
}

// ---------------------------------------------------------------------------
// Kernel 1: per-modality encoders + z + mean + initial fusion MLP.
// Block = 256 threads = 4 nodes x 64 dims. h[n,m,32] staged in LDS.
// ---------------------------------------------------------------------------
__global__ void k_encode(const float* __restrict__ data,
                         const float* __restrict__ w1, const float* __restrict__ b1,
                         const float* __restrict__ w2, const float* __restrict__ b2,
                         const float* __restrict__ fw, const float* __restrict__ fb,
                         float* __restrict__ zout,  // d_out + 6N, [N,4,64]
                         float* __restrict__ x0,    // [N,64]
                         int N) {
    __shared__ float hsh[4][MMOD][32];
    __shared__ float zmean[4][DLAT];
    const int t = threadIdx.x;
    const int nb = blockIdx.x * 4;
    for (int idx = t; idx < 512; idx += 256) {
        const int nl = idx >> 7, m = (idx >> 5) & 3, j = idx & 31;
        const int n = nb + nl;
        float v = 0.0f;
        if (n < N) {
            const float xm = data[n * MMOD + m];
            v = gelu_exact(xm * w1[m * 32 + j] + b1[m * 32 + j]);
        }
        hsh[nl][m][j] = v;
    }
    __syncthreads();
    const int nl = t >> 6, d = t & 63;
    const int n = nb + nl;
    float zm = 0.0f;
    if (n < N) {
        #pragma unroll
        for (int m = 0; m < MMOD; ++m) {
            float acc = b2[m * DLAT + d];
            const float* w = w2 + m * DLAT * 32 + d * 32;
            #pragma unroll
            for (int j = 0; j < 32; ++j) acc += hsh[nl][m][j] * w[j];
            zout[n * (MMOD * DLAT) + m * DLAT + d] = acc;
            zm += acc;
        }
    }
    zmean[nl][d] = zm * 0.25f;
    __syncthreads();
    if (n < N) {
        float acc = fb[d];
        const float* w = fw + d * DLAT;
        #pragma unroll
        for (int dp = 0; dp < DLAT; ++dp) acc += zmean[nl][dp] * w[dp];
        x0[n * DLAT + d] = acc;
    }
}

// ---------------------------------------------------------------------------
// Kernel: reset per-layer accumulators (accum=0, denom=0, max=-FLT_MAX enc).
// ---------------------------------------------------------------------------
__global__ void k_init(float* __restrict__ accum, float* __restrict__ denom,
                       unsigned* __restrict__ mxe, int N) {
    const int i = blockIdx.x * blockDim.x + threadIdx.x;
    if (i < N * OD) accum[i] = 0.0f;
    if (i < N * NHEAD) {
        denom[i] = 0.0f;
        mxe[i] = fenc(-3.402823466e38f);
    }
}

// ---------------------------------------------------------------------------
// Kernel: out[n,o] = sum_d x[n,d]*w[o,d] + b[o]  via V_WMMA_F32_16X16X4_F32.
// One wave per 16-node tile; loops over all 16 column tiles (OD=256).
// Requires N % 16 == 0 (N=20000 -> 1250 tiles). Wave-uniform guard keeps
// EXEC all-ones for WMMA.
// ---------------------------------------------------------------------------
__global__ void k_project(const float* __restrict__ x, const float* __restrict__ w,
                          const float* __restrict__ b, float* __restrict__ out,
                          int ntiles) {
    const int wave = (blockIdx.x * blockDim.x + threadIdx.x) >> 5;
    const int lane = threadIdx.x & 31;
    if (wave >= ntiles) return; // wave-uniform branch
    const int n0 = wave * 16;
    const int r = lane & 15;   // A row / B column within tile
    const int kh = lane >> 4;  // K-pair selector (ISA A/B layout)
    const float* xrow = x + (n0 + r) * DLAT + 2 * kh;
    for (int ot = 0; ot < 16; ++ot) {
        const int o0 = ot * 16;
        const float bias = b[o0 + r]; // D: col = lane&15 in both halves
        v8f c;
        #pragma unroll
        for (int v = 0; v < 8; ++v) c[v] = bias;
        const float* wrow = w + (o0 + r) * DLAT + 2 * kh;
        #pragma unroll
        for (int k = 0; k < 16; ++k) {
            v2f a, bb;
            a[0] = xrow[k * 4];     // VGPR0: K = 4k + 2*kh
            a[1] = xrow[k * 4 + 1]; // VGPR1: K = 4k + 2*kh + 1
            bb[0] = wrow[k * 4];
            bb[1] = wrow[k * 4 + 1];
            c = __builtin_amdgcn_wmma_f32_16x16x4_f32(
                    false, a, false, bb, (short)0, c, false, false);
        }
        // D layout: lane<16 -> M=v ; lane>=16 -> M=v+8 ; N=lane&15
        float* orow = out + (n0 + 8 * kh) * OD + o0 + r;
        #pragma unroll
        for (int v = 0; v < 8; ++v) orow[v * OD] = c[v];
    }
}

// ---------------------------------------------------------------------------
// Kernel: edge logits (LeakyReLU + att dot) and segment max (encoded atomicMax).
// One thread per (edge, head). Self-loops are edges e >= E.
// ---------------------------------------------------------------------------
__global__ void k_logits(const long long* __restrict__ srcp,
                         const long long* __restrict__ dstp,
                         const float* __restrict__ xl, const float* __restrict__ xr,
                         const float* __restrict__ att,  // layer-offset [4,64]
                         float* __restrict__ logit, unsigned* __restrict__ mxe,
                         int E, int Et) {
    const int t = blockIdx.x * blockDim.x + threadIdx.x;
    if (t >= Et * NHEAD) return;
    const int e = t >> 2, h = t & 3;
    int s, d;
    if (e < E) { s = (int)srcp[e]; d = (int)dstp[e]; }
    else       { s = e - E; d = e - E; }
    const float* xls = xl + (long)s * OD + h * DLAT;
    const float* xrd = xr + (long)d * OD + h * DLAT;
    const float* a  = att + h * DLAT;
    float acc = 0.0f;
    #pragma unroll 8
    for (int i = 0; i < DLAT; ++i) {
        float v = xls[i] + xrd[i];
        v = (v > 0.0f) ? v : 0.2f * v; // leaky_relu 0.2
        acc += a[i] * v;
    }
    logit[t] = acc;
    atomicMax(&mxe[d * NHEAD + h], fenc(acc));
}

// ---------------------------------------------------------------------------
// Kernel: alpha numerators + scatter-accumulate  accum[dst] += ex * xl[src].
// ---------------------------------------------------------------------------
__global__ void k_alpha(const long long* __restrict__ srcp,
                        const long long* __restrict__ dstp,
                        const float* __restrict__ xl,
                        const float* __restrict__ logit,
                        const unsigned* __restrict__ mxe,
                        float* __restrict__ denom, float* __restrict__ accum,
                        int E, int Et) {
    const int t = blockIdx.x * blockDim.x + threadIdx.x;
    if (t >= Et * NHEAD) return;
    const int e = t >> 2, h = t & 3;
    int s, d;
    if (e < E) { s = (int)srcp[e]; d = (int)dstp[e]; }
    else       { s = e - E; d = e - E; }
    const float ex = expf(logit[t] - fdec(mxe[d * NHEAD + h]));
    atomicAdd(&denom[d * NHEAD + h], ex);
    const float* xls = xl + (long)s * OD + h * DLAT;
    float* acc = accum + (long)d * OD + h * DLAT;
    #pragma unroll 8
    for (int i = 0; i < DLAT; ++i) atomicAdd(&acc[i], ex * xls[i]);
}

// ---------------------------------------------------------------------------
// Kernel: x_next[n,d] = gelu( mean_h(accum/denom) + bias[d] )
// ---------------------------------------------------------------------------
__global__ void k_finish(const float* __restrict__ accum,
                         const float* __restrict__ denom,
                         const float* __restrict__ bias,  // layer-offset [64]
                         float* __restrict__ xnext, int N) {
    const int i = blockIdx.x * blockDim.x + threadIdx.x;
    if (i >= N * DLAT) return;
    const int n = i >> 6, d = i & 63;
    float o = 0.0f;
    #pragma unroll
    for (int h = 0; h < NHEAD; ++h)
        o += accum[n * OD + h * DLAT + d] / (denom[n * NHEAD + h] + 1e-16f);
    xnext[i] = gelu_exact(o * 0.25f + bias[d]);
}

// ---------------------------------------------------------------------------
// Kernel: interpretable fusion + risk/unc heads. One 64-thread block per node.
// Small weights (q_w/k_w ~80 KB) are L2-resident.
// ---------------------------------------------------------------------------
__global__ void k_head(const float* __restrict__ x, const float* __restrict__ zin,
                       const float* __restrict__ qw, const float* __restrict__ qb,
                       const float* __restrict__ kw, const float* __restrict__ kb,
                       const float* __restrict__ rw1, const float* __restrict__ rb1,
                       const float* __restrict__ rw2, const float* __restrict__ rb2,
                       const float* __restrict__ uw1, const float* __restrict__ ub1,
                       const float* __restrict__ uw2, const float* __restrict__ ub2,
                       float* __restrict__ out_risk, float* __restrict__ out_unc,
                       float* __restrict__ out_attn, int N) {
    __shared__ float xs[DLAT], q[DLAT], keys[MMOD][DLAT], zs[MMOD][DLAT];
    __shared__ float sc[MMOD], at[MMOD], fused[DLAT], hr[32], hu[32];
    const int n = blockIdx.x;
    const int t = threadIdx.x;
    if (n >= N) return;
    xs[t] = x[n * DLAT + t];
    #pragma unroll
    for (int m = 0; m < MMOD; ++m) zs[m][t] = zin[n * (MMOD * DLAT) + m * DLAT + t];
    __syncthreads();
    { // query
        float acc = qb[t];
        const float* w = qw + t * DLAT;
        #pragma unroll
        for (int d = 0; d < DLAT; ++d) acc += xs[d] * w[d];
        q[t] = acc;
    }
    #pragma unroll
    for (int m = 0; m < MMOD; ++m) { // keys[n,m,e] = sum_d z[n,m,d]*kw[m,e,d]
        float acc = kb[m * DLAT + t];
        const float* w = kw + m * DLAT * DLAT + t * DLAT;
        #pragma unroll
        for (int d = 0; d < DLAT; ++d) acc += zs[m][d] * w[d];
        keys[m][t] = acc;
    }
    __syncthreads();
    if (t < MMOD) {
        float s = 0.0f;
        for (int e2 = 0; e2 < DLAT; ++e2) s += q[e2] * keys[t][e2];
        sc[t] = s * 0.125f; // 1/sqrt(64)
    }
    __syncthreads();
    if (t == 0) {
        float m0 = fmaxf(fmaxf(sc[0], sc[1]), fmaxf(sc[2], sc[3]));
        float e0 = expf(sc[0] - m0), e1 = expf(sc[1] - m0);
        float e2 = expf(sc[2] - m0), e3 = expf(sc[3] - m0);
        const float inv = 1.0f / (e0 + e1 + e2 + e3);
        at[0] = e0 * inv; at[1] = e1 * inv; at[2] = e2 * inv; at[3] = e3 * inv;
        #pragma unroll
        for (int m = 0; m < MMOD; ++m) out_attn[n * MMOD + m] = at[m];
    }
    __syncthreads();
    {
        float f = 0.0f;
        #pragma unroll
        for (int m = 0; m < MMOD; ++m) f += at[m] * zs[m][t];
        fused[t] = f;
    }
    __syncthreads();
    if (t < 32) {
        float a = rb1[t], u = ub1[t];
        const float* wr = rw1 + t * DLAT;
        const float* wu = uw1 + t * DLAT;
        #pragma unroll
        for (int d = 0; d < DLAT; ++d) { a += fused[d] * wr[d]; u += fused[d] * wu[d]; }
        hr[t] = gelu_exact(a);
        hu[t] = gelu_exact(u);
    }
    __syncthreads();
    if (t == 0) {
        float r = rb2[0], u = ub2[0];
        #pragma unroll
        for (int i = 0; i < 32; ++i) { r += hr[i] * rw2[i]; u += hu[i] * uw2[i]; }
        out_risk[n] = r;
        out_unc[n]  = 1.0f / (1.0f + expf(-u));
    }
}

// ---------------------------------------------------------------------------
extern "C" void kernel_launch(void* const* d_in, const int* in_sizes, int n_in,
                              void* d_out, int out_size, void* d_ws, size_t ws_size,
                              hipStream_t stream) {
    const int N = in_sizes[0] / MMOD;       // 20000
    const int E = in_sizes[1] / 2;          // 320000
    const int Et = E + N;                   // with self loops

    // ---- inputs (setup_inputs order) ----
    const float* data   = (const float*)d_in[0];
    const long long* ei = (const long long*)d_in[1]; // int64 edge_index [2,E]
    const float* enc_w1 = (const float*)d_in[2];
    const float* enc_b1 = (const float*)d_in[3];
    const float* enc_w2 = (const float*)d_in[4];
    const float* enc_b2 = (const float*)d_in[5];
    const float* fus_w  = (const float*)d_in[6];
    const float* fus_b  = (const float*)d_in[7];
    const float* gat_wl = (const float*)d_in[8];
    const float* gat_bl = (const float*)d_in[9];
    const float* gat_wr = (const float*)d_in[10];
    const float* gat_br = (const float*)d_in[11];
    const float* gat_att  = (const float*)d_in[12];
    const float* gat_bias = (const float*)d_in[13];
    const float* q_w = (const float*)d_in[14];
    const float* q_b = (const float*)d_in[15];
    const float* k_w = (const float*)d_in[16];
    const float* k_b = (const float*)d_in[17];
    const float* r_w1 = (const float*)d_in[18];
    const float* r_b1 = (const float*)d_in[19];
    const float* r_w2 = (const float*)d_in[20];
    const float* r_b2 = (const float*)d_in[21];
    const float* u_w1 = (const float*)d_in[22];
    const float* u_b1 = (const float*)d_in[23];
    const float* u_w2 = (const float*)d_in[24];
    const float* u_b2 = (const float*)d_in[25];

    const long long* srcp = ei;
    const long long* dstp = ei + E;

    // ---- outputs: risk[N] | unc[N] | attn[N*4] | z[N*4*64] ----
    float* out  = (float*)d_out;
    float* o_risk = out;
    float* o_unc  = out + N;
    float* o_attn = out + 2 * N;
    float* o_z    = out + 2 * N + N * MMOD;

    // ---- workspace layout (floats), ~78 MB total, fits L2 ----
    float* ws = (float*)d_ws;
    float* xa    = ws;                  // [N,64]
    float* xb    = xa + (size_t)N * DLAT;      // [N,64]
    float* xl    = xb + (size_t)N * DLAT;      // [N,256]
    float* xr    = xl + (size_t)N * OD;        // [N,256]
    float* accum = xr + (size_t)N * OD;        // [N,256]
    float* logit = accum + (size_t)N * OD;     // [Et,4]
    float* denom = logit + (size_t)Et * NHEAD; // [N,4]
    unsigned* mxe = (unsigned*)(denom + (size_t)N * NHEAD); // [N,4]

    // 1) encoders + initial fusion
    k_encode<<<(N + 3) / 4, 256, 0, stream>>>(data, enc_w1, enc_b1, enc_w2, enc_b2,
                                              fus_w, fus_b, o_z, xa, N);

    const int ntiles = N / 16; // N % 16 == 0 for this problem
    const int projBlocks = (ntiles * 32 + 255) / 256;
    const int edgeThreads = Et * NHEAD;
    const int edgeBlocks = (edgeThreads + 255) / 256;
    const int ndBlocks = (N * DLAT + 255) / 256;
    const int initBlocks = (N * OD + 255) / 256;

    float* xcur = xa;
    float* xnext = xb;
    for (int l = 0; l < NLAY; ++l) {
        const float* wl = gat_wl + (size_t)l * OD * DLAT;
        const float* bl = gat_bl + (size_t)l * OD;
        const float* wr = gat_wr + (size_t)l * OD * DLAT;
        const float* br = gat_br + (size_t)l * OD;
        const float* att  = gat_att + (size_t)l * NHEAD * DLAT;
        const float* bias = gat_bias + (size_t)l * DLAT;

        k_init<<<initBlocks, 256, 0, stream>>>(accum, denom, mxe, N);
        k_project<<<projBlocks, 256, 0, stream>>>(xcur, wl, bl, xl, ntiles);
        k_project<<<projBlocks, 256, 0, stream>>>(xcur, wr, br, xr, ntiles);
        k_logits<<<edgeBlocks, 256, 0, stream>>>(srcp, dstp, xl, xr, att,
                                                 logit, mxe, E, Et);
        k_alpha<<<edgeBlocks, 256, 0, stream>>>(srcp, dstp, xl, logit, mxe,
                                                denom, accum, E, Et);
        k_finish<<<ndBlocks, 256, 0, stream>>>(accum, denom, bias, xnext, N);

        float* tmp = xcur; xcur = xnext; xnext = tmp;
    }

    // 3) interpretable fusion + heads
    k_head<<<N, 64, 0, stream>>>(xcur, o_z, q_w, q_b, k_w, k_b,
                                 r_w1, r_b1, r_w2, r_b2,
                                 u_w1, u_b1, u_w2, u_b2,
                                 o_risk, o_unc, o_attn, N);
    (void)n_in; (void)out_size; (void)ws_size;
}